// RPEAttention_73426760892660
// MI455X (gfx1250) — compile-verified
//
#include <hip/hip_runtime.h>

typedef __attribute__((ext_vector_type(2))) float v2f;
typedef __attribute__((ext_vector_type(8))) float v8f;

#define NTOK 512
#define DIMC 128
#define HEADS 4
#define DH 32
#define INDC 5
#define RPE_SCALE 0.17677669529663687f /* 32^-0.5 */

// D = A(16x4,f32) * B(4x16,f32) + C(16x16,f32)
// A/B lane layout (ISA 7.12.2): lane half = lane>>4, K = 2*half + vgpr
// C/D layout: m = r + 8*half, n = lane&15
static __device__ __forceinline__ v8f wmma_f32(v2f a, v2f b, v8f c) {
  return __builtin_amdgcn_wmma_f32_16x16x4_f32(false, a, false, b, (short)0, c,
                                               false, false);
}

// ---------------- Kernel 0: qkv = w_qkv(384x128) @ x(128x512) ----------------
__global__ __launch_bounds__(128) void qkv_kernel(
    const float* __restrict__ x, const float* __restrict__ w_qkv,
    float* __restrict__ qkv) {
  const int lane = threadIdx.x & 31;
  const int wave = threadIdx.x >> 5;
  const int gw = blockIdx.x * 4 + wave;  // 0..767 tiles (24 x 32)
  const int Mt = (gw >> 5) * 16;
  const int Nt = (gw & 31) * 16;
  const int half = lane >> 4;
  const int l16 = lane & 15;
  v8f acc = {};
  const int arow = Mt + l16;
  const int bcol = Nt + l16;
  for (int kb = 0; kb < DIMC; kb += 4) {
    const int k0 = kb + 2 * half;
    v2f a = *(const v2f*)(w_qkv + arow * DIMC + k0);  // contiguous in K
    v2f b;
    b.x = x[(k0 + 0) * NTOK + bcol];
    b.y = x[(k0 + 1) * NTOK + bcol];
    acc = wmma_f32(a, b, acc);
  }
#pragma unroll
  for (int r = 0; r < 8; ++r) {
    const int m = r + 8 * half;
    qkv[(Mt + m) * NTOK + bcol] = acc[r];
  }
}

// ------------- Kernel 1: fused RPE attention for one (head, I-tile) ----------
__global__ __launch_bounds__(128) void attn_kernel(
    const float* __restrict__ qkv, const float* __restrict__ indicator,
    const float* __restrict__ w_ind, float* __restrict__ attnout) {
  __shared__ float sBq[INDC][NTOK];  // 10 KB
  __shared__ float sAk[INDC][16];
  __shared__ float sM[INDC][INDC];
  __shared__ float sSim[16][NTOK];  // 32 KB
  __shared__ float sRed[16][8];
  __shared__ float sPv[2][8][32];  // [d-tile][r][lane] cross-wave reduce

  const int tid = threadIdx.x;
  const int lane = tid & 31;
  const int wave = tid >> 5;
  const int h = blockIdx.x & 3;
  const int It = (blockIdx.x >> 2) * 16;
  const int half = lane >> 4;
  const int l16 = lane & 15;

  const float* q = qkv + (h * DH) * NTOK;            // q[d][i]
  const float* kM = qkv + (DIMC + h * DH) * NTOK;    // k[d][j]
  const float* vM = qkv + (2 * DIMC + h * DH) * NTOK;  // v[d][j]
  const float* wq = w_ind + (h * DH) * INDC;           // ind_q weights
  const float* wk = w_ind + (DIMC + h * DH) * INDC;    // ind_k weights

  // ---- precompute Bq (5x512), Ak (5x16), M (5x5) into LDS ----
  for (int idx = tid; idx < INDC * NTOK; idx += 128) {
    const int c = idx / NTOK, j = idx - c * NTOK;
    float s = 0.f;
    for (int d = 0; d < DH; ++d) s += kM[d * NTOK + j] * wq[d * INDC + c];
    sBq[c][j] = s;
  }
  for (int idx = tid; idx < INDC * 16; idx += 128) {
    const int c = idx >> 4, m = idx & 15;
    float s = 0.f;
    for (int d = 0; d < DH; ++d) s += q[d * NTOK + It + m] * wk[d * INDC + c];
    sAk[c][m] = s;
  }
  if (tid < INDC * INDC) {
    const int c = tid / INDC, cp = tid - c * INDC;
    float s = 0.f;
    for (int d = 0; d < DH; ++d) s += wq[d * INDC + c] * wk[d * INDC + cp];
    sM[c][cp] = s;
  }
  __syncthreads();

  // hoist the q A-fragments across the full K=32
  v2f qa[8];
#pragma unroll
  for (int s = 0; s < 8; ++s) {
    const int k0 = 4 * s + 2 * half;
    qa[s].x = q[(k0 + 0) * NTOK + It + l16];
    qa[s].y = q[(k0 + 1) * NTOK + It + l16];
  }

  // ---- sim strip: 8 J-tiles per wave ----
  for (int t = 0; t < 8; ++t) {
    const int Jt = (wave + 4 * t) * 16;
    v8f acc = {};
#pragma unroll
    for (int s = 0; s < 8; ++s) {
      const int k0 = 4 * s + 2 * half;
      v2f b;
      b.x = kM[(k0 + 0) * NTOK + Jt + l16];
      b.y = kM[(k0 + 1) * NTOK + Jt + l16];
      acc = wmma_f32(qa[s], b, acc);
    }
    const int j = Jt + l16;
    if (t < 7) {  // stream next tile's indicator into cache
      const int jn = (wave + 4 * (t + 1)) * 16 + l16;
      __builtin_prefetch(indicator + (size_t)It * NTOK + jn, 0, 1);
    }
#pragma unroll
    for (int r = 0; r < 8; ++r) {
      const int m = r + 8 * half;
      const int i = It + m;
      float tv[INDC];
#pragma unroll
      for (int c = 0; c < INDC; ++c)
        tv[c] = indicator[((size_t)c * NTOK + i) * NTOK + j];
      float corr = 0.f;
#pragma unroll
      for (int cp = 0; cp < INDC; ++cp) {
        float u = sAk[cp][m] + sBq[cp][j];
#pragma unroll
        for (int c = 0; c < INDC; ++c) u += sM[c][cp] * tv[c];
        corr += u * tv[cp];
      }
      sSim[m][Jt + l16] = RPE_SCALE * (acc[r] + corr);
    }
  }
  __syncthreads();

  // ---- softmax over j (16 rows x 512; 8 threads per row) ----
  const int row = tid >> 3;
  const int seg = tid & 7;
  const int c0 = seg * 64;
  float mx = -3.402823466e38f;
  for (int c = 0; c < 64; ++c) mx = fmaxf(mx, sSim[row][c0 + c]);
  sRed[row][seg] = mx;
  __syncthreads();
  float rowmax = sRed[row][0];
  for (int s = 1; s < 8; ++s) rowmax = fmaxf(rowmax, sRed[row][s]);
  __syncthreads();
  float sum = 0.f;
  for (int c = 0; c < 64; ++c) {
    const float e = __expf(sSim[row][c0 + c] - rowmax);
    sSim[row][c0 + c] = e;
    sum += e;
  }
  sRed[row][seg] = sum;
  __syncthreads();
  float rowsum = 0.f;
  for (int s = 0; s < 8; ++s) rowsum += sRed[row][s];
  const float inv = 1.f / rowsum;
  for (int c = 0; c < 64; ++c) sSim[row][c0 + c] *= inv;
  __syncthreads();

  // ---- out[i][d] = attn(16x512) @ v^T(512x32); K split over 4 waves ----
  const int ntile = wave & 1;  // which 16 of the 32 d's
  const int khalf = wave >> 1; // which 256 of K
  const int dcol = ntile * 16 + l16;
  v8f pv = {};
  for (int s = 0; s < 64; ++s) {
    const int jb = khalf * 256 + 4 * s + 2 * half;
    v2f a = *(const v2f*)(&sSim[l16][jb]);        // attn fragment
    v2f b = *(const v2f*)(vM + dcol * NTOK + jb); // v^T fragment (contig in j)
    pv = wmma_f32(a, b, pv);
  }
  if (khalf == 1) {
#pragma unroll
    for (int r = 0; r < 8; ++r) sPv[ntile][r][lane] = pv[r];
  }
  __syncthreads();
  if (khalf == 0) {
#pragma unroll
    for (int r = 0; r < 8; ++r) {
      const float val = pv[r] + sPv[ntile][r][lane];
      const int m = r + 8 * half;  // local i
      attnout[(h * DH + dcol) * NTOK + It + m] = val;
    }
  }
}

// ---- Kernel 2: out = w_out(128x128) @ attnout(128x512) + b_out ----
__global__ __launch_bounds__(128) void proj_kernel(
    const float* __restrict__ attnout, const float* __restrict__ w_out,
    const float* __restrict__ b_out, float* __restrict__ out) {
  const int lane = threadIdx.x & 31;
  const int wave = threadIdx.x >> 5;
  const int gw = blockIdx.x * 4 + wave;  // 0..255 tiles (8 x 32)
  const int Mt = (gw >> 5) * 16;
  const int Nt = (gw & 31) * 16;
  const int half = lane >> 4;
  const int l16 = lane & 15;
  v8f acc = {};
  const int arow = Mt + l16;
  const int bcol = Nt + l16;
  for (int kb = 0; kb < DIMC; kb += 4) {
    const int k0 = kb + 2 * half;
    v2f a = *(const v2f*)(w_out + arow * DIMC + k0);
    v2f b;
    b.x = attnout[(k0 + 0) * NTOK + bcol];
    b.y = attnout[(k0 + 1) * NTOK + bcol];
    acc = wmma_f32(a, b, acc);
  }
#pragma unroll
  for (int r = 0; r < 8; ++r) {
    const int m = r + 8 * half;
    out[(Mt + m) * NTOK + bcol] = acc[r] + b_out[Mt + m];
  }
}

extern "C" void kernel_launch(void* const* d_in, const int* in_sizes, int n_in,
                              void* d_out, int out_size, void* d_ws,
                              size_t ws_size, hipStream_t stream) {
  (void)in_sizes; (void)n_in; (void)out_size; (void)ws_size;
  const float* x = (const float*)d_in[0];          // (1,128,512)
  const float* indicator = (const float*)d_in[1];  // (1,5,512,512)
  const float* w_qkv = (const float*)d_in[2];      // (384,128)
  const float* w_ind = (const float*)d_in[3];      // (256,5)
  const float* w_out = (const float*)d_in[4];      // (128,128)
  const float* b_out = (const float*)d_in[5];      // (128,)
  float* out = (float*)d_out;                      // (1,128,512)

  float* qkv = (float*)d_ws;                 // 384*512 f32 = 786 KB
  float* attnout = qkv + 3 * DIMC * NTOK;    // 128*512 f32 = 262 KB

  qkv_kernel<<<192, 128, 0, stream>>>(x, w_qkv, qkv);
  attn_kernel<<<HEADS * (NTOK / 16), 128, 0, stream>>>(qkv, indicator, w_ind,
                                                       attnout);
  proj_kernel<<<64, 128, 0, stream>>>(attnout, w_out, b_out, out);
}